// RNN_Plant_36421322670153
// MI455X (gfx1250) — compile-verified
//
#include <hip/hip_runtime.h>
#include <math.h>

typedef __attribute__((ext_vector_type(2))) float v2f;
typedef __attribute__((ext_vector_type(8))) float v8f;

#define T_STEPS 600
#define BATCH   512
#define HID     256
#define NIN     7
#define ROWS    16               // batch rows per workgroup
#define NWG     (BATCH / ROWS)   // 32 workgroups
#define NTHREADS 256             // 8 waves (wave32)
#define KTOT    264              // 256 hidden + 7 scaled inputs + 1 zero pad
#define KSTEPS  (KTOT / 4)       // 66 WMMA K-steps (f32 16x16x4)
#define HS_STRIDE 268            // 264 cols + 4 pad floats -> bank spread
#define NTILES  16               // 256 output cols / 16
#define TAU_F   10.0f

// ---- dynamic LDS layout (bytes) ----
#define WFRAG_ELEMS (NTILES * KSTEPS * 32)            // v2f fragments of [W_rec;W_in;0]
#define OFF_WFRAG 0
#define OFF_RED   (OFF_WFRAG + WFRAG_ELEMS * 8)       // 270336 : 32 doubles
#define OFF_HS    (OFF_RED + 32 * 8)                  // 270592 : 16*268 floats
#define OFF_BSUM  (OFF_HS + ROWS * HS_STRIDE * 4)     // 287744 : 256 floats
#define OFF_WO    (OFF_BSUM + HID * 4)                // 288768 : 512 floats
#define OFF_PART  (OFF_WO + HID * 2 * 4)              // 290816 : 16*2*8 floats
#define SMEM_BYTES (OFF_PART + ROWS * 2 * 8 * 4)      // 291840 total ( < 320KB WGP LDS )

__launch_bounds__(NTHREADS)
__global__ void rnn_plant_kernel(
    const float* __restrict__ state_init,
    const float* __restrict__ hidden_init,
    const float* __restrict__ go_seq,
    const float* __restrict__ target_seq_t,
    const float* __restrict__ target_go_seq,
    const float* __restrict__ mask,
    const float* __restrict__ W_in,
    const float* __restrict__ b_in,
    const float* __restrict__ W_rec,
    const float* __restrict__ b_rec,
    const float* __restrict__ W_out,
    const float* __restrict__ b_out,
    double* __restrict__ partials)
{
  extern __shared__ char smem[];
  v2f*    Wfrag = (v2f*)(smem + OFF_WFRAG);
  double* red   = (double*)(smem + OFF_RED);
  float*  Hs    = (float*)(smem + OFF_HS);
  float*  bsum  = (float*)(smem + OFF_BSUM);
  float*  Wo    = (float*)(smem + OFF_WO);
  float*  part  = (float*)(smem + OFF_PART);

  const int tid  = threadIdx.x;
  const int wg   = blockIdx.x;
  const int lane = tid & 31;
  const int wave = tid >> 5;
  const int half = lane >> 4;   // which 16-lane half of the wave
  const int nloc = lane & 15;

  // ---- Stage extended B operand [W_rec ; W_in ; zero-row] into LDS in WMMA
  //      fragment order: fragment f = ((tile*KSTEPS)+k)*32 + lane, where
  //      v2f.x = row (4k + 2*half), v2f.y = row (4k + 1 + 2*half), col = tile*16 + lane%16
  for (int f = tid; f < WFRAG_ELEMS; f += NTHREADS) {
    int L    = f & 31;
    int k    = (f >> 5) % KSTEPS;
    int tile = f / (KSTEPS * 32);
    int h2   = L >> 4;
    int n    = tile * 16 + (L & 15);
    int r0   = 4 * k + 2 * h2;
    int r1   = r0 + 1;
    float x = (r0 < HID) ? W_rec[r0 * HID + n] : ((r0 - HID) < NIN ? W_in[(r0 - HID) * HID + n] : 0.0f);
    float y = (r1 < HID) ? W_rec[r1 * HID + n] : ((r1 - HID) < NIN ? W_in[(r1 - HID) * HID + n] : 0.0f);
    v2f b; b.x = x; b.y = y;
    Wfrag[f] = b;
  }
  if (tid < HID) bsum[tid] = b_in[tid] + b_rec[tid];
  for (int i = tid; i < HID * 2; i += NTHREADS) Wo[i] = W_out[i];

  // ---- hidden state: register copy (C-layout positions) + LDS copy (A layout source)
  float hreg[2][8];
#pragma unroll
  for (int tt = 0; tt < 2; ++tt) {
    int n = (2 * wave + tt) * 16 + nloc;
#pragma unroll
    for (int v = 0; v < 8; ++v) {
      int m = v + 8 * half;
      float h = hidden_init[(wg * ROWS + m) * HID + n];
      hreg[tt][v] = h;
      Hs[m * HS_STRIDE + n] = h;
    }
  }

  // ---- plant state + first-step scaled input columns
  float pos0 = 0.f, pos1 = 0.f, vel0 = 0.f, vel1 = 0.f;
  double l1acc = 0.0, l2acc = 0.0;
  if (tid < ROWS) {
    int brow = wg * ROWS + tid;
    pos0 = state_init[brow * 4 + 0];
    pos1 = state_init[brow * 4 + 1];
    vel0 = state_init[brow * 4 + 2];
    vel1 = state_init[brow * 4 + 3];
    float g   = go_seq[brow];
    float ti0 = target_seq_t[brow * 2 + 0];
    float ti1 = target_seq_t[brow * 2 + 1];
    float* row = &Hs[tid * HS_STRIDE + 256];
    row[0] = g   * 0.25f;  row[1] = ti0 * 0.25f; row[2] = ti1 * 0.25f;
    row[3] = pos0 * 0.25f; row[4] = pos1 * 0.25f;
    row[5] = vel0 * 40.0f; row[6] = vel1 * 40.0f; row[7] = 0.0f;
  }
  __syncthreads();

  const v2f* fragA = &Wfrag[(2 * wave + 0) * (KSTEPS * 32) + lane];
  const v2f* fragB = &Wfrag[(2 * wave + 1) * (KSTEPS * 32) + lane];
  const float* arow = &Hs[(lane & 15) * HS_STRIDE + 2 * half];

  for (int t = 0; t < T_STEPS; ++t) {
    // Early global loads for the scalar tail (hidden behind WMMA latency)
    float mk = 0.f, goL = 0.f, tg0 = 0.f, tg1 = 0.f, gN = 0.f, tiN0 = 0.f, tiN1 = 0.f;
    if (tid < ROWS) {
      int brow = wg * ROWS + tid;
      mk  = mask[t];
      goL = go_seq[t * BATCH + brow];
      tg0 = target_go_seq[(t * BATCH + brow) * 2 + 0];
      tg1 = target_go_seq[(t * BATCH + brow) * 2 + 1];
      int tn = (t + 1 < T_STEPS) ? (t + 1) : t;
      gN   = go_seq[tn * BATCH + brow];
      tiN0 = target_seq_t[(tn * BATCH + brow) * 2 + 0];
      tiN1 = target_seq_t[(tn * BATCH + brow) * 2 + 1];
    }

    // ---- activity = [hidden | inp] @ [W_rec ; W_in]  via f32 WMMA, 2 tiles/wave
    v8f acc00 = {}; v8f acc01 = {}; v8f acc10 = {}; v8f acc11 = {};
#pragma unroll
    for (int k = 0; k < KSTEPS; ++k) {
      v2f a  = *(const v2f*)&arow[4 * k];
      v2f bA = fragA[k * 32];
      v2f bB = fragB[k * 32];
      if (k & 1) {
        acc01 = __builtin_amdgcn_wmma_f32_16x16x4_f32(false, a, false, bA, (short)0, acc01, false, false);
        acc11 = __builtin_amdgcn_wmma_f32_16x16x4_f32(false, a, false, bB, (short)0, acc11, false, false);
      } else {
        acc00 = __builtin_amdgcn_wmma_f32_16x16x4_f32(false, a, false, bA, (short)0, acc00, false, false);
        acc10 = __builtin_amdgcn_wmma_f32_16x16x4_f32(false, a, false, bB, (short)0, acc10, false, false);
      }
    }

    // ---- hidden EMA update in registers
#pragma unroll
    for (int tt = 0; tt < 2; ++tt) {
      int n = (2 * wave + tt) * 16 + nloc;
      float bs = bsum[n];
      v8f s = (tt == 0) ? (acc00 + acc01) : (acc10 + acc11);
#pragma unroll
      for (int v = 0; v < 8; ++v)
        hreg[tt][v] = 0.9f * hreg[tt][v] + 0.1f * (s[v] + bs);
    }
    __syncthreads();  // all WMMA A-reads of old hidden done

#pragma unroll
    for (int tt = 0; tt < 2; ++tt) {
      int n = (2 * wave + tt) * 16 + nloc;
#pragma unroll
      for (int v = 0; v < 8; ++v)
        Hs[(v + 8 * half) * HS_STRIDE + n] = hreg[tt][v];
    }
    __syncthreads();  // new hidden visible

    // ---- action = hidden @ W_out : 256 threads -> (row m, out j, 8 segs of 32)
    {
      int m = tid >> 4;
      int j = (tid >> 3) & 1;
      int seg = tid & 7;
      const float* hrow = &Hs[m * HS_STRIDE + seg * 32];
      const float* wcol = &Wo[seg * 32 * 2 + j];
      float p = 0.f;
#pragma unroll
      for (int i = 0; i < 32; ++i) p = fmaf(hrow[i], wcol[2 * i], p);
      part[(m * 2 + j) * 8 + seg] = p;
    }
    __syncthreads();

    // ---- plant update + online loss (one owner thread per batch row)
    if (tid < ROWS) {
      float a0 = b_out[0], a1 = b_out[1];
#pragma unroll
      for (int s = 0; s < 8; ++s) {
        a0 += part[(tid * 2 + 0) * 8 + s];
        a1 += part[(tid * 2 + 1) * 8 + s];
      }
      float ac0 = a0 - 1.0e-4f * pos0 - 1.0e-3f * vel0;
      float ac1 = a1 - 1.0e-4f * pos1 - 1.0e-3f * vel1;
      vel0 += ac0; vel1 += ac1;
      pos0 += vel0; pos1 += vel1;

      float d0 = pos0 - tg0, d1 = pos1 - tg1;
      float vt0 = vel0 * TAU_F, vt1 = vel1 * TAU_F;
      float e0 = 1.0f - expf(-(d0 * d0) - 2.0f * vt0 * vt0) + (ac0 * 100.0f) * (ac0 * 100.0f);
      float e1 = 1.0f - expf(-(d1 * d1) - 2.0f * vt1 * vt1) + (ac1 * 100.0f) * (ac1 * 100.0f);
      l1acc += (double)(mk * (e0 + e1));
      l2acc += (double)(mk * 10.0f * goL * 0.5f * (vt0 * vt0 + vt1 * vt1));

      // scaled input columns for step t+1
      float* row = &Hs[tid * HS_STRIDE + 256];
      row[0] = gN   * 0.25f; row[1] = tiN0 * 0.25f; row[2] = tiN1 * 0.25f;
      row[3] = pos0 * 0.25f; row[4] = pos1 * 0.25f;
      row[5] = vel0 * 40.0f; row[6] = vel1 * 40.0f;
    }
    __syncthreads();  // inp + hidden stable for next iteration
  }

  // ---- deterministic in-block reduction -> per-WG partials
  if (tid < ROWS) { red[tid] = l1acc; red[ROWS + tid] = l2acc; }
  __syncthreads();
  if (tid == 0) {
    double s1 = 0.0, s2 = 0.0;
    for (int i = 0; i < ROWS; ++i) { s1 += red[i]; s2 += red[ROWS + i]; }
    partials[wg * 2 + 0] = s1;
    partials[wg * 2 + 1] = s2;
  }
}

__global__ void finalize_kernel(const double* __restrict__ partials, float* __restrict__ out) {
  if (threadIdx.x == 0 && blockIdx.x == 0) {
    double s1 = 0.0, s2 = 0.0;
    for (int i = 0; i < NWG; ++i) { s1 += partials[2 * i]; s2 += partials[2 * i + 1]; }
    double loss = s1 / ((double)T_STEPS * BATCH * 2) + s2 / ((double)T_STEPS * BATCH);
    out[0] = (float)loss;
  }
}

extern "C" void kernel_launch(void* const* d_in, const int* in_sizes, int n_in,
                              void* d_out, int out_size, void* d_ws, size_t ws_size,
                              hipStream_t stream) {
  (void)in_sizes; (void)n_in; (void)out_size; (void)ws_size;
  const float* state_init    = (const float*)d_in[0];
  const float* hidden_init   = (const float*)d_in[1];
  const float* go_seq        = (const float*)d_in[2];
  const float* target_seq_t  = (const float*)d_in[3];
  const float* target_go_seq = (const float*)d_in[4];
  const float* mask          = (const float*)d_in[5];
  const float* W_in          = (const float*)d_in[6];
  const float* b_in          = (const float*)d_in[7];
  const float* W_rec         = (const float*)d_in[8];
  const float* b_rec         = (const float*)d_in[9];
  const float* W_out         = (const float*)d_in[10];
  const float* b_out         = (const float*)d_in[11];
  double* partials = (double*)d_ws;

  // Opt in to >64KB dynamic LDS (gfx1250 WGP supports 320KB per workgroup)
  static_assert(SMEM_BYTES <= 320 * 1024, "exceeds WGP LDS");
  (void)hipFuncSetAttribute((const void*)rnn_plant_kernel,
                            hipFuncAttributeMaxDynamicSharedMemorySize, SMEM_BYTES);

  rnn_plant_kernel<<<NWG, NTHREADS, SMEM_BYTES, stream>>>(
      state_init, hidden_init, go_seq, target_seq_t, target_go_seq, mask,
      W_in, b_in, W_rec, b_rec, W_out, b_out, partials);
  finalize_kernel<<<1, 32, 0, stream>>>(partials, (float*)d_out);
}